// YoloLoss_54288386621894
// MI455X (gfx1250) — compile-verified
//
#include <hip/hip_runtime.h>
#include <cmath>

#define EPSF        1e-7f
#define IOU_THR     0.5f
#define PEN_F       0.1f
#define BOX_W       5.0f
#define SMALL_SCALE 2.0f
#define SMALL_THR   0.05f
#define MAX_NG      128
#define MAX_C       96        // LDS tile capacity per row (C=80 in reference)
#define FOUR_OVER_PI2 0.4052847345693511f

typedef float v2f __attribute__((ext_vector_type(2)));
typedef float v8f __attribute__((ext_vector_type(8)));

#if __has_builtin(__builtin_amdgcn_global_load_async_to_lds_b128) && \
    __has_builtin(__builtin_amdgcn_s_wait_asynccnt)
#define HAVE_ASYNC_LDS 1
typedef int v4i __attribute__((vector_size(16)));
typedef __attribute__((address_space(1))) v4i* g_v4i_p;   // global int4*
typedef __attribute__((address_space(3))) v4i* l_v4i_p;   // LDS int4*
#endif

__device__ __forceinline__ float wred_f(float v) {
#pragma unroll
    for (int o = 16; o > 0; o >>= 1) v += __shfl_down(v, o, 32);
    return v;
}
__device__ __forceinline__ int wred_i(int v) {
#pragma unroll
    for (int o = 16; o > 0; o >>= 1) v += __shfl_down(v, o, 32);
    return v;
}

// ---------------------------------------------------------------- zero init
__global__ void zero_ws_kernel(unsigned* w, int n) {
    int t = blockIdx.x * blockDim.x + threadIdx.x;
    if (t < n) w[t] = 0u;
}

// ------------------------------------------------- kernel A: match + fused losses
// acc[b*4+0]=loc_sum, +1=obj_sum, +2=pen_sum, +3=cls_sum (kernel B)
// cnt[b*2+0]=n_match, +1=n_valid
__global__ void iou_match_kernel(const float* __restrict__ y_hat,
                                 const float* __restrict__ y,
                                 const float* __restrict__ obj,
                                 const int*   __restrict__ ctgt,
                                 const int*   __restrict__ min_obj,
                                 float* __restrict__ acc,
                                 int*   __restrict__ cnt,
                                 int*   __restrict__ out_tgt,
                                 float* __restrict__ out_wgt,
                                 int Np, int Ng) {
    // fields: 0=x1 1=y1 2=x2 3=y2 4=area 5=w_small 6..7 pad
    __shared__ float gl[MAX_NG][8];

    const int id = blockIdx.x * blockDim.x + threadIdx.x;
    const int b  = (blockIdx.x * blockDim.x) / Np;   // block never crosses a batch

    if (threadIdx.x < Ng) {
        const int g = threadIdx.x;
        float4 gb = reinterpret_cast<const float4*>(y)[(size_t)b * Ng + g];
        float gx1 = gb.x - gb.z * 0.5f, gy1 = gb.y - gb.w * 0.5f;
        float gx2 = gb.x + gb.z * 0.5f, gy2 = gb.y + gb.w * 0.5f;
        gl[g][0] = gx1; gl[g][1] = gy1; gl[g][2] = gx2; gl[g][3] = gy2;
        gl[g][4] = (gx2 - gx1) * (gy2 - gy1);
        gl[g][5] = ((gb.z < SMALL_THR) || (gb.w < SMALL_THR)) ? SMALL_SCALE : 1.0f;
    }
    __syncthreads();

    float4 ph = reinterpret_cast<const float4*>(y_hat)[id];
    const float px1 = ph.x - ph.z * 0.5f, py1 = ph.y - ph.w * 0.5f;
    const float px2 = ph.x + ph.z * 0.5f, py2 = ph.y + ph.w * 0.5f;
    const float area_p = (px2 - px1) * (py2 - py1);

    float best = -INFINITY;
    int   bidx = 0;
    for (int g = 0; g < Ng; ++g) {
        float iw = fminf(px2, gl[g][2]) - fmaxf(px1, gl[g][0]);
        float ih = fminf(py2, gl[g][3]) - fmaxf(py1, gl[g][1]);
        iw = fmaxf(iw, 0.0f); ih = fmaxf(ih, 0.0f);
        float inter = iw * ih;
        float iou = inter / (area_p + gl[g][4] - inter);   // no eps (matches iou_mat)
        if (iou > best) { best = iou; bidx = g; }          // first-max like jnp.argmax
    }

    const float s  = obj[id];
    const float mo = (float)min_obj[0];
    const bool valid   = s > mo;
    const bool matched = valid && (best > IOU_THR);
    const float wsm = gl[bidx][5];

    float loc_c = 0.0f;
    if (matched) {
        const float gx1 = gl[bidx][0], gy1 = gl[bidx][1];
        const float gx2 = gl[bidx][2], gy2 = gl[bidx][3];
        const float w1 = px2 - px1, h1 = py2 - py1;
        const float w2 = gx2 - gx1, h2 = gy2 - gy1;
        float iw = fmaxf(fminf(px2, gx2) - fmaxf(px1, gx1), 0.0f);
        float ih = fmaxf(fminf(py2, gy2) - fmaxf(py1, gy1), 0.0f);
        float inter = iw * ih;
        float uni   = w1 * h1 + w2 * h2 - inter + EPSF;
        float iou2  = inter / uni;
        float cw = fmaxf(px2, gx2) - fminf(px1, gx1);
        float ch = fmaxf(py2, gy2) - fminf(py1, gy1);
        float c2 = cw * cw + ch * ch + EPSF;
        float dx = (px1 + px2) * 0.5f - (gx1 + gx2) * 0.5f;
        float dy = (py1 + py2) * 0.5f - (gy1 + gy2) * 0.5f;
        float cd2 = dx * dx + dy * dy;
        float da = atanf(w1 / (h1 + EPSF)) - atanf(w2 / (h2 + EPSF));
        float v  = FOUR_OVER_PI2 * da * da;
        float alpha = v / (1.0f - iou2 + v + EPSF);
        float ciou  = iou2 - (cd2 / c2 + alpha * v);
        loc_c = (1.0f - ciou) * wsm;
    }

    const float lbl  = matched ? 1.0f : 0.0f;
    const float bce  = fmaxf(s, 0.0f) - s * lbl + log1pf(expf(-fabsf(s)));
    const float wobj = (matched && (s < 0.5f)) ? 2.0f : 1.0f;
    float obj_c = valid ? bce * wobj : 0.0f;
    float pen_c = (valid && !matched) ? s : 0.0f;

    out_tgt[id] = ctgt[b * Ng + bidx];
    out_wgt[id] = matched ? wsm : 0.0f;

    loc_c = wred_f(loc_c);
    obj_c = wred_f(obj_c);
    pen_c = wred_f(pen_c);
    int nm = wred_i(matched ? 1 : 0);
    int nv = wred_i(valid ? 1 : 0);
    if ((threadIdx.x & 31) == 0) {
        atomicAdd(&acc[b * 4 + 0], loc_c);
        atomicAdd(&acc[b * 4 + 1], obj_c);
        atomicAdd(&acc[b * 4 + 2], pen_c);
        atomicAdd(&cnt[b * 2 + 0], nm);
        atomicAdd(&cnt[b * 2 + 1], nv);
    }
}

// ------------------------------------------------- kernel B: CE via LDS tile + WMMA
// One wave owns 16 consecutive prediction rows (one contiguous 16*C-float chunk).
// Stage tile into LDS (async DMA when available), then:
//   rowmax from LDS, sumexp via V_WMMA_F32_16X16X4_F32 with B=ones, gather from LDS.
__global__ void ce_wmma_kernel(const float* __restrict__ cls_pred,
                               const int*   __restrict__ tgt,
                               const float* __restrict__ wgt,
                               float* __restrict__ acc,
                               int Np, int C, int total_rows) {
    __shared__ float tileLDS[4][16 * MAX_C];
    __shared__ float se_lds[4][16];

    const int lane = threadIdx.x & 31;
    const int wid  = threadIdx.x >> 5;                     // 0..3 (block=128)
    const int row0_orig = (blockIdx.x * 4 + wid) * 16;
    int row0 = row0_orig;
    if (row0 + 16 > total_rows) row0 = total_rows - 16;    // clamp; keep EXEC all-1s
    const bool live = (row0_orig + (lane & 15)) < total_rows;

    // ---- stage 16*C contiguous floats into this wave's LDS region ----
    const float* gsrc = cls_pred + (size_t)row0 * C;
    float*       ldst = &tileLDS[wid][0];
    const int nvec4 = (16 * C) >> 2;                       // 16B transfers
#ifdef HAVE_ASYNC_LDS
    for (int t = lane; t < nvec4; t += 32) {
        __builtin_amdgcn_global_load_async_to_lds_b128(
            (g_v4i_p)(unsigned long long)(gsrc + 4 * t),
            (l_v4i_p)(unsigned)(unsigned long long)(ldst + 4 * t),
            0, 0);
    }
    __builtin_amdgcn_s_wait_asynccnt(0);
    __syncthreads();
#else
    for (int t = lane; t < nvec4; t += 32) {
        reinterpret_cast<float4*>(ldst)[t] =
            reinterpret_cast<const float4*>(gsrc)[t];
    }
    __syncthreads();
#endif

    const float* rowbase = ldst + (lane & 15) * C;

    // row max (both half-waves compute the same row's max) — vectorized LDS reads
    float m = -INFINITY;
    for (int c4 = 0; c4 < (C >> 2); ++c4) {
        float4 x = reinterpret_cast<const float4*>(rowbase)[c4];
        m = fmaxf(m, fmaxf(fmaxf(x.x, x.y), fmaxf(x.z, x.w)));
    }

    // A-layout: VGPR0 = K{0|2}, VGPR1 = K{1|3}; hi half-wave carries K=2,3
    const int koff = (lane < 16) ? 0 : 2;
    v8f d = {};
    v2f onesv; onesv.x = 1.0f; onesv.y = 1.0f;
    for (int j = 0; j < C; j += 4) {
        float2 xv = *reinterpret_cast<const float2*>(rowbase + j + koff);
        v2f a;
        a.x = expf(xv.x - m);
        a.y = expf(xv.y - m);
        d = __builtin_amdgcn_wmma_f32_16x16x4_f32(
                /*neg_a=*/false, a, /*neg_b=*/false, onesv,
                /*c_mod=*/(short)0, d, /*reuse_a=*/false, /*reuse_b=*/false);
    }

    // D layout: lane 0 VGPR v -> M=v ; lane 16 VGPR v -> M=v+8 (all columns equal)
    if (lane == 0) {
#pragma unroll
        for (int v = 0; v < 8; ++v) se_lds[wid][v] = d[v];
    }
    if (lane == 16) {
#pragma unroll
        for (int v = 0; v < 8; ++v) se_lds[wid][8 + v] = d[v];
    }
    __syncthreads();

    float contrib = 0.0f;
    if (lane < 16) {
        float se = se_lds[wid][lane];
        int   myrow = row0 + lane;
        int   t  = tgt[myrow];
        float pv = rowbase[t];                             // gather from LDS tile
        float ce = -(pv - m - logf(se));
        contrib = live ? ce * wgt[myrow] : 0.0f;
    }
    contrib = wred_f(contrib);
    if (lane == 0) {
        int b = row0 / Np;                                 // 16 | Np, never crosses batch
        atomicAdd(&acc[b * 4 + 3], contrib);
    }
}

// ------------------------------------------------- kernel C: final combine
__global__ void final_kernel(const float* __restrict__ acc,
                             const int*   __restrict__ cnt,
                             float* __restrict__ out, int B) {
    if (threadIdx.x == 0 && blockIdx.x == 0) {
        float tot = 0.0f; int nvb = 0;
        for (int b = 0; b < B; ++b) {
            int nm = cnt[b * 2 + 0], nv = cnt[b * 2 + 1];
            float loc  = (nm > 0) ? acc[b * 4 + 0] / (float)nm : 0.0f;
            float objl = (nv > 0) ? acc[b * 4 + 1] / (float)nv : 0.0f;
            float pen  = PEN_F * acc[b * 4 + 2];
            float cls  = (nm > 0) ? acc[b * 4 + 3] / (float)nm : 0.0f;
            float pb = BOX_W * loc + objl + pen + cls;
            if (nv > 0) { tot += pb; ++nvb; }
        }
        out[0] = tot / (float)(nvb > 0 ? nvb : 1);
    }
}

extern "C" void kernel_launch(void* const* d_in, const int* in_sizes, int n_in,
                              void* d_out, int out_size, void* d_ws, size_t ws_size,
                              hipStream_t stream) {
    const float* y_hat = (const float*)d_in[0];   // (B,Np,4)
    const float* y     = (const float*)d_in[1];   // (B,Ng,4)
    const float* obj   = (const float*)d_in[2];   // (B,Np)
    const float* cpred = (const float*)d_in[3];   // (B,Np,C)
    const int*   ctgt  = (const int*)d_in[4];     // (B,Ng)
    const int*   mobj  = (const int*)d_in[5];     // scalar

    const int B  = 16;
    const int Np = in_sizes[2] / B;
    const int Ng = in_sizes[4] / B;
    const int C  = in_sizes[3] / in_sizes[2];
    const int total = B * Np;

    float* acc     = (float*)d_ws;                 // 4*B floats
    int*   cnt     = (int*)(acc + 4 * B);          // 2*B ints
    int*   out_tgt = (int*)(cnt + 2 * B);          // total ints
    float* out_wgt = (float*)(out_tgt + total);    // total floats

    zero_ws_kernel<<<1, 256, 0, stream>>>((unsigned*)d_ws, 6 * B);
    iou_match_kernel<<<total / 256, 256, 0, stream>>>(
        y_hat, y, obj, ctgt, mobj, acc, cnt, out_tgt, out_wgt, Np, Ng);
    ce_wmma_kernel<<<(total + 63) / 64, 128, 0, stream>>>(
        cpred, out_tgt, out_wgt, acc, Np, C, total);
    final_kernel<<<1, 1, 0, stream>>>(acc, cnt, (float*)d_out, B);
}